// GaussianMixtureAdam_44349832298694
// MI455X (gfx1250) — compile-verified
//
#include <hip/hip_runtime.h>
#include <hip/hip_bf16.h>
#include <math.h>

#define DDIM   64
#define KCOMP  32
#define REGEPS 1e-6f
#define LOG2PI 1.8378770664093453f
#define STRD   68   // LDS row stride (floats), conflict-free for b64 fragment loads
#define PTS_PER_BLOCK 128   // 4 waves x 32 points

typedef __attribute__((ext_vector_type(2))) float v2f;
typedef __attribute__((ext_vector_type(8))) float v8f;

// ---------------------------------------------------------------------------
// Phase 1: per-component Sigma = L L^T + eps I, Cholesky, triangular inverse,
// and the per-component additive constant. One block per component.
// ---------------------------------------------------------------------------
__global__ __launch_bounds__(256) void gmm_prep_kernel(
    const float* __restrict__ cov,      // (K, D, D)
    const float* __restrict__ weights,  // (K,)
    float* __restrict__ Ainv,           // (K, D, D) row-major, Lc^{-1} (lower)
    float* __restrict__ constk)         // (K,)
{
  __shared__ float S[DDIM][DDIM];   // sigma -> in-place Cholesky (lower = Lc)
  __shared__ float L[DDIM][DDIM];   // raw tril factor, then reused for Lc^{-1}

  const int k = blockIdx.x;
  const int t = threadIdx.x;
  const float* C = cov + (size_t)k * DDIM * DDIM;

  // L = tril(C) + eps*I
  for (int idx = t; idx < DDIM * DDIM; idx += 256) {
    const int r = idx / DDIM, c = idx % DDIM;
    float v = (c <= r) ? C[idx] : 0.0f;
    if (r == c) v += REGEPS;
    L[r][c] = v;
  }
  __syncthreads();

  // S = L L^T + eps*I
  for (int idx = t; idx < DDIM * DDIM; idx += 256) {
    const int r = idx / DDIM, c = idx % DDIM;
    float acc = (r == c) ? REGEPS : 0.0f;
    const int m = (r < c) ? r : c;
    for (int j = 0; j <= m; ++j) acc = fmaf(L[r][j], L[c][j], acc);
    S[r][c] = acc;
  }
  __syncthreads();

  // Right-looking Cholesky: lower triangle of S becomes Lc.
  for (int j = 0; j < DDIM; ++j) {
    if (t == 0) S[j][j] = sqrtf(S[j][j]);
    __syncthreads();
    const float dinv = 1.0f / S[j][j];
    for (int i = j + 1 + t; i < DDIM; i += 256) S[i][j] *= dinv;
    __syncthreads();
    for (int idx = t; idx < DDIM * DDIM; idx += 256) {
      const int r = idx / DDIM, c = idx % DDIM;
      if (c > j && r >= c) S[r][c] = fmaf(-S[r][j], S[c][j], S[r][c]);
    }
    __syncthreads();
  }

  // Invert Lc (forward substitution, one column per thread) into L.
  if (t < DDIM) {
    const int c = t;
    for (int i = 0; i < c; ++i) L[i][c] = 0.0f;
    L[c][c] = 1.0f / S[c][c];
    for (int i = c + 1; i < DDIM; ++i) {
      float acc = 0.0f;
      for (int j = c; j < i; ++j) acc = fmaf(S[i][j], L[j][c], acc);
      L[i][c] = -acc / S[i][i];
    }
  }
  __syncthreads();

  for (int idx = t; idx < DDIM * DDIM; idx += 256)
    Ainv[(size_t)k * DDIM * DDIM + idx] = L[idx / DDIM][idx % DDIM];

  if (t == 0) {
    float logdet = 0.0f;
    for (int j = 0; j < DDIM; ++j) logdet += logf(S[j][j]);
    constk[k] = logf(weights[k]) - logdet - 0.5f * (float)DDIM * LOG2PI;
  }
}

// ---------------------------------------------------------------------------
// Phase 2: maha(k,n) via WMMA F32 16x16x4 and streaming logsumexp over k.
// 128 threads (4 wave32) per block, 128 points per block, 32 points per wave
// (two 16-row A tiles per wave -> every B fragment feeds 2 WMMAs).
// ---------------------------------------------------------------------------
__global__ __launch_bounds__(128) void gmm_logprob_kernel(
    const float* __restrict__ X,       // (N, D)
    const float* __restrict__ Ainv,    // (K, D, D)
    const float* __restrict__ means,   // (K, D)
    const float* __restrict__ constk,  // (K,)
    float* __restrict__ out)           // (N,)
{
  __shared__ float Xs[PTS_PER_BLOCK][STRD];  // 128-point tile of X
  __shared__ float As[DDIM][STRD];           // current A_k, row=dout, col=kdin
  __shared__ float Mu[DDIM];
  __shared__ float Ck[KCOMP];
  __shared__ float Maha[4][2][16];           // [wave][row-tile][row]

  const int tid  = threadIdx.x;
  const int wave = tid >> 5;
  const int lane = tid & 31;
  const int half = lane >> 4;   // lane group: selects which K the VGPR holds
  const int l16  = lane & 15;
  const int n0   = blockIdx.x * PTS_PER_BLOCK;

  if (tid < KCOMP) Ck[tid] = constk[tid];

  // Stage X tile (float4-vectorized, coalesced).
  for (int idx = tid; idx < PTS_PER_BLOCK * (DDIM / 4); idx += 128) {
    const int r = idx >> 4, c4 = idx & 15;
    const float4 v = ((const float4*)(X + (size_t)(n0 + r) * DDIM))[c4];
    Xs[r][c4 * 4 + 0] = v.x; Xs[r][c4 * 4 + 1] = v.y;
    Xs[r][c4 * 4 + 2] = v.z; Xs[r][c4 * 4 + 3] = v.w;
  }
  __syncthreads();

  // Per-lane X fragments for the wave's two 16-point row tiles.
  // A-matrix 16x4 f32 layout: v0 -> K = 4s + 2*half, v1 -> K = 4s + 2*half + 1.
  const int wp = wave * 32 + l16;
  v2f xf[2][16];
#pragma unroll
  for (int s = 0; s < 16; ++s) {
    xf[0][s] = *(const v2f*)&Xs[wp][s * 4 + 2 * half];
    xf[1][s] = *(const v2f*)&Xs[wp + 16][s * 4 + 2 * half];
  }

  float lse_m = -INFINITY;  // one point per lane: n0 + wave*32 + lane
  float lse_s = 0.0f;

  for (int k = 0; k < KCOMP; ++k) {
    __syncthreads();
    // Stage A_k and mu_k.
    for (int idx = tid; idx < DDIM * (DDIM / 4); idx += 128) {
      const int r = idx >> 4, c4 = idx & 15;
      const float4 v =
          ((const float4*)(Ainv + (size_t)k * DDIM * DDIM + r * DDIM))[c4];
      As[r][c4 * 4 + 0] = v.x; As[r][c4 * 4 + 1] = v.y;
      As[r][c4 * 4 + 2] = v.z; As[r][c4 * 4 + 3] = v.w;
    }
    if (tid < DDIM) Mu[tid] = means[k * DDIM + tid];
    if (k + 1 < KCOMP)  // warm L2/L1 for next component (global_prefetch_b8)
      __builtin_prefetch(Ainv + (size_t)(k + 1) * DDIM * DDIM + tid * 32, 0, 1);
    __syncthreads();

    v8f acc[2][4];
#pragma unroll
    for (int t2 = 0; t2 < 2; ++t2)
#pragma unroll
      for (int db = 0; db < 4; ++db) acc[t2][db] = (v8f){};

#pragma unroll
    for (int s = 0; s < 16; ++s) {
      const v2f mu2 = *(const v2f*)&Mu[s * 4 + 2 * half];
      const v2f a0 = xf[0][s] - mu2;  // diff fragments, row tiles 0/1
      const v2f a1 = xf[1][s] - mu2;
#pragma unroll
      for (int db = 0; db < 4; ++db) {
        // B-matrix 4x16 f32: v0 lane = B[K=4s+2*half, N=l16] = A[db*16+l16][4s+2*half]
        const v2f bf = *(const v2f*)&As[db * 16 + l16][s * 4 + 2 * half];
        acc[0][db] = __builtin_amdgcn_wmma_f32_16x16x4_f32(
            false, a0, false, bf, (short)0, acc[0][db], false, false);
        acc[1][db] = __builtin_amdgcn_wmma_f32_16x16x4_f32(
            false, a1, false, bf, (short)0, acc[1][db], false, false);
      }
    }

    // Square-accumulate over the 4 dim-blocks, then reduce over N
    // (16 lanes within each half). VGPR j of a tile holds row M = j + 8*half.
#pragma unroll
    for (int t2 = 0; t2 < 2; ++t2) {
#pragma unroll
      for (int j = 0; j < 8; ++j) {
        float v = 0.0f;
#pragma unroll
        for (int db = 0; db < 4; ++db)
          v = fmaf(acc[t2][db][j], acc[t2][db][j], v);
        v += __shfl_xor(v, 1, 32);
        v += __shfl_xor(v, 2, 32);
        v += __shfl_xor(v, 4, 32);
        v += __shfl_xor(v, 8, 32);
        if (l16 == 0) Maha[wave][t2][j + 8 * half] = v;
      }
    }
    __syncthreads();

    // Streaming logsumexp; lane owns point n0 + wave*32 + lane.
    {
      const float lp = Ck[k] - 0.5f * Maha[wave][half][l16];
      if (lp > lse_m) {
        lse_s = lse_s * expf(lse_m - lp) + 1.0f;
        lse_m = lp;
      } else {
        lse_s += expf(lp - lse_m);
      }
    }
  }

  out[n0 + wave * 32 + lane] = lse_m + logf(lse_s);
}

extern "C" void kernel_launch(void* const* d_in, const int* in_sizes, int n_in,
                              void* d_out, int out_size, void* d_ws, size_t ws_size,
                              hipStream_t stream) {
  const float* X       = (const float*)d_in[0];
  const float* weights = (const float*)d_in[1];
  const float* means   = (const float*)d_in[2];
  const float* cov     = (const float*)d_in[3];
  float* out = (float*)d_out;

  const int N = in_sizes[0] / DDIM;

  float* Ainv   = (float*)d_ws;                               // K*D*D floats
  float* constk = (float*)((char*)d_ws + (size_t)KCOMP * DDIM * DDIM * sizeof(float));

  gmm_prep_kernel<<<KCOMP, 256, 0, stream>>>(cov, weights, Ainv, constk);
  gmm_logprob_kernel<<<N / PTS_PER_BLOCK, 128, 0, stream>>>(X, Ainv, means, constk, out);
  (void)n_in; (void)out_size; (void)ws_size;
}